// StVenantKirchhoffShellEnergy_48180943127064
// MI455X (gfx1250) — compile-verified
//
#include <hip/hip_runtime.h>

// ---------------------------------------------------------------------------
// St. Venant-Kirchhoff shell energy (membrane + bending + drill stabilization)
// Latency-bound gather workload (inputs ~44MB -> L2-resident on 192MB L2).
// Strategy: grid-stride over elements, compile-time-unrolled batch loop so all
// 12 float4 DOF gathers issue as one clause, f32 register accumulation,
// v_wmma_f32_16x16x4_f32 wave reduction, deterministic 2-stage global sum.
// ---------------------------------------------------------------------------

typedef float v2f __attribute__((ext_vector_type(2)));
typedef float v8f __attribute__((ext_vector_type(8)));

// Physics constants: computed in double (as Python does) then narrowed to f32.
static constexpr double EMOD_d = 100000.0;
static constexpr double NU_d   = 0.3;
static constexpr double TH_d   = 0.01;
static constexpr float  kMU    = (float)(EMOD_d / (2.0 * (1.0 + NU_d)));
static constexpr float  kLMBDA = (float)(EMOD_d * NU_d / ((1.0 + NU_d) * (1.0 - 2.0 * NU_d)));
static constexpr float  kDBEND = (float)(EMOD_d * TH_d * TH_d * TH_d / (12.0 * (1.0 - NU_d * NU_d)));
static constexpr float  kDRILL = (float)(0.0001 * EMOD_d * TH_d * TH_d * TH_d);
static constexpr float  kTH    = (float)TH_d;
static constexpr float  kNU    = (float)NU_d;

// Wave32 sum reduction. WMMA path: A = 16x4 ones, B holds one lane-partial per
// (VGPR0,lane) slot (other half zero). D[m][n] = colsum(B)[n] for every m, so
// the sum of ALL 256 D entries equals 16 * (sum of all B entries) = 16 * wave
// total -- layout-independent, full f32 precision.
__device__ __forceinline__ float wave_reduce_sum(float acc) {
#if defined(__gfx1250__) && __has_builtin(__builtin_amdgcn_wmma_f32_16x16x4_f32)
  v2f a; a[0] = 1.0f; a[1] = 1.0f;     // A = all ones (both A VGPRs, all lanes)
  v2f b; b[0] = acc;  b[1] = 0.0f;     // one partial per lane in B VGPR0
  v8f c = {};
  c = __builtin_amdgcn_wmma_f32_16x16x4_f32(
      /*neg_a=*/false, a, /*neg_b=*/false, b,
      /*c_mod=*/(short)0, c, /*reuse_a=*/false, /*reuse_b=*/false);
  float s = ((c[0] + c[1]) + (c[2] + c[3])) + ((c[4] + c[5]) + (c[6] + c[7]));
  #pragma unroll
  for (int off = 16; off > 0; off >>= 1) s += __shfl_xor(s, off, 32);
  return s * (1.0f / 16.0f);
#else
  #pragma unroll
  for (int off = 16; off > 0; off >>= 1) acc += __shfl_xor(acc, off, 32);
  return acc;
#endif
}

// Per-batch energy given DOF float4s (disp = .x/.y, rot = .z/.w) and the
// element's shape-function gradients.
__device__ __forceinline__ float element_batch_energy(
    const float4& q0, const float4& q1, const float4& q2,
    float d0x, float d0y, float g1x, float g1y, float g2x, float g2y,
    float det) {
  // Displacement gradient H = ed^T * dN_dx ; F = I + H
  const float Hxx = q0.x * d0x + q1.x * g1x + q2.x * g2x;
  const float Hxy = q0.x * d0y + q1.x * g1y + q2.x * g2y;
  const float Hyx = q0.y * d0x + q1.y * g1x + q2.y * g2x;
  const float Hyy = q0.y * d0y + q1.y * g1y + q2.y * g2y;
  const float F00 = 1.0f + Hxx, F01 = Hxy;
  const float F10 = Hyx,        F11 = 1.0f + Hyy;

  // C = F^T F ; E = 0.5 (C - I)
  const float C00 = F00 * F00 + F10 * F10;
  const float C01 = F00 * F01 + F10 * F11;
  const float C11 = F01 * F01 + F11 * F11;
  const float E00 = 0.5f * (C00 - 1.0f);
  const float E11 = 0.5f * (C11 - 1.0f);
  const float E01 = 0.5f * C01;
  const float trE  = E00 + E11;
  const float trEE = E00 * E00 + E11 * E11 + 2.0f * E01 * E01;
  const float membrane = (0.5f * kLMBDA * trE * trE + kMU * trEE) * det * kTH;

  // Curvature K = er^T * dN_dx (rot = q.z, q.w)
  const float Kxx = q0.z * d0x + q1.z * g1x + q2.z * g2x;
  const float Kxy = q0.z * d0y + q1.z * g1y + q2.z * g2y;
  const float Kyx = q0.w * d0x + q1.w * g1x + q2.w * g2x;
  const float Kyy = q0.w * d0y + q1.w * g1y + q2.w * g2y;
  const float kxy = 0.5f * (Kxy + Kyx);
  const float bending =
      kDBEND * (Kxx * Kxx + Kyy * Kyy + 2.0f * kNU * Kxx * Kyy +
                2.0f * (1.0f - kNU) * kxy * kxy) * det;

  const float stab = kDRILL * (q0.z * q0.z + q1.z * q1.z + q2.z * q2.z);

  return membrane + bending + stab;
}

template <int BATCH>
__global__ __launch_bounds__(256) void shell_energy_partial_t(
    const float* __restrict__ u, const float* __restrict__ coords,
    const long long* __restrict__ elems, int num_elems, int num_nodes,
    float* __restrict__ partials) {
  const int tid    = blockIdx.x * blockDim.x + threadIdx.x;
  const int stride = gridDim.x * blockDim.x;
  const float2* __restrict__ co = (const float2*)coords;

  float acc = 0.0f;
  for (int e = tid; e < num_elems; e += stride) {
    const long long* ep = elems + 3ll * (long long)e;
    // Speculative prefetch of this thread's next element row
    // (global_prefetch_b8; OOB translation failures silently dropped).
    __builtin_prefetch(ep + 3ll * (long long)stride, 0, 0);

    // 3 x i64 indices as one b128 + one b64 load.
    const longlong2 i01 = *(const longlong2*)ep;
    const int i0 = (int)i01.x;
    const int i1 = (int)i01.y;
    const int i2 = (int)ep[2];

    // Issue ALL gathers up front: 3 coord float2 + 3*BATCH DOF float4.
    const float2 c0 = co[i0], c1 = co[i1], c2 = co[i2];
    float4 q[BATCH][3];
    #pragma unroll
    for (int b = 0; b < BATCH; ++b) {
      const float4* __restrict__ ub =
          (const float4*)(u + (size_t)b * 4u * (size_t)num_nodes);
      q[b][0] = ub[i0];
      q[b][1] = ub[i1];
      q[b][2] = ub[i2];
    }

    // Geometry (overlaps with outstanding DOF gathers).
    // J = coords^T * dN_dxi with dN_dxi = [[-1,-1],[1,0],[0,1]]
    const float jxx = c1.x - c0.x;   // J00
    const float jxy = c2.x - c0.x;   // J01
    const float jyx = c1.y - c0.y;   // J10
    const float jyy = c2.y - c0.y;   // J11
    const float det  = jxx * jyy - jxy * jyx;
    const float rdet = 1.0f / det;
    // invJ rows; dN_dx rows: node0 = -(g1+g2), node1 = g1, node2 = g2
    const float g1x =  jyy * rdet, g1y = -jxy * rdet;
    const float g2x = -jyx * rdet, g2y =  jxx * rdet;
    const float d0x = -(g1x + g2x), d0y = -(g1y + g2y);

    #pragma unroll
    for (int b = 0; b < BATCH; ++b) {
      acc += element_batch_energy(q[b][0], q[b][1], q[b][2],
                                  d0x, d0y, g1x, g1y, g2x, g2y, det);
    }
  }

  // Uniform here: all 256 threads active, EXEC all-ones -> WMMA legal.
  const float wave_total = wave_reduce_sum(acc);

  __shared__ float wsum[32];
  const int lane = threadIdx.x & 31;
  const int wv   = threadIdx.x >> 5;
  if (lane == 0) wsum[wv] = wave_total;
  __syncthreads();
  if (threadIdx.x == 0) {
    float s = 0.0f;
    const int nw = (blockDim.x + 31) >> 5;
    for (int i = 0; i < nw; ++i) s += wsum[i];
    partials[blockIdx.x] = s;
  }
}

// Generic fallback for unexpected batch counts.
__global__ __launch_bounds__(256) void shell_energy_partial_gen(
    const float* __restrict__ u, const float* __restrict__ coords,
    const long long* __restrict__ elems, int num_elems, int num_nodes,
    int batch, float* __restrict__ partials) {
  const int tid    = blockIdx.x * blockDim.x + threadIdx.x;
  const int stride = gridDim.x * blockDim.x;
  const float2* __restrict__ co = (const float2*)coords;

  float acc = 0.0f;
  for (int e = tid; e < num_elems; e += stride) {
    const long long* ep = elems + 3ll * (long long)e;
    const longlong2 i01 = *(const longlong2*)ep;
    const int i0 = (int)i01.x;
    const int i1 = (int)i01.y;
    const int i2 = (int)ep[2];

    const float2 c0 = co[i0], c1 = co[i1], c2 = co[i2];
    const float jxx = c1.x - c0.x, jxy = c2.x - c0.x;
    const float jyx = c1.y - c0.y, jyy = c2.y - c0.y;
    const float det  = jxx * jyy - jxy * jyx;
    const float rdet = 1.0f / det;
    const float g1x =  jyy * rdet, g1y = -jxy * rdet;
    const float g2x = -jyx * rdet, g2y =  jxx * rdet;
    const float d0x = -(g1x + g2x), d0y = -(g1y + g2y);

    for (int b = 0; b < batch; ++b) {
      const float4* __restrict__ ub =
          (const float4*)(u + (size_t)b * 4u * (size_t)num_nodes);
      acc += element_batch_energy(ub[i0], ub[i1], ub[i2],
                                  d0x, d0y, g1x, g1y, g2x, g2y, det);
    }
  }

  const float wave_total = wave_reduce_sum(acc);

  __shared__ float wsum[32];
  const int lane = threadIdx.x & 31;
  const int wv   = threadIdx.x >> 5;
  if (lane == 0) wsum[wv] = wave_total;
  __syncthreads();
  if (threadIdx.x == 0) {
    float s = 0.0f;
    const int nw = (blockDim.x + 31) >> 5;
    for (int i = 0; i < nw; ++i) s += wsum[i];
    partials[blockIdx.x] = s;
  }
}

__global__ __launch_bounds__(256) void shell_energy_finalize(
    const float* __restrict__ partials, int n, float scale,
    float* __restrict__ out) {
  float s = 0.0f;
  for (int i = threadIdx.x; i < n; i += blockDim.x) s += partials[i];

  const float wave_total = wave_reduce_sum(s);

  __shared__ float wsum[32];
  const int lane = threadIdx.x & 31;
  const int wv   = threadIdx.x >> 5;
  if (lane == 0) wsum[wv] = wave_total;
  __syncthreads();
  if (threadIdx.x == 0) {
    float t = 0.0f;
    const int nw = (blockDim.x + 31) >> 5;
    for (int i = 0; i < nw; ++i) t += wsum[i];
    out[0] = t * scale;
  }
}

extern "C" void kernel_launch(void* const* d_in, const int* in_sizes, int n_in,
                              void* d_out, int out_size, void* d_ws, size_t ws_size,
                              hipStream_t stream) {
  const float*     u      = (const float*)d_in[0];       // (batch, 4*num_nodes) f32
  const float*     coords = (const float*)d_in[1];       // (num_nodes, 2) f32
  const long long* elems  = (const long long*)d_in[2];   // (num_elems, 3) i64

  const int num_nodes = in_sizes[1] / 2;
  const int num_elems = in_sizes[2] / 3;
  const int batch     = (num_nodes > 0) ? in_sizes[0] / (4 * num_nodes) : 1;

  float* partials = (float*)d_ws;
  int nblocks = 1024;                                    // ~4 elems/thread
  const int max_blocks = (int)(ws_size / sizeof(float));
  if (nblocks > max_blocks) nblocks = max_blocks;
  if (nblocks < 1) nblocks = 1;

  if (batch == 4) {
    shell_energy_partial_t<4><<<nblocks, 256, 0, stream>>>(
        u, coords, elems, num_elems, num_nodes, partials);
  } else {
    shell_energy_partial_gen<<<nblocks, 256, 0, stream>>>(
        u, coords, elems, num_elems, num_nodes, batch, partials);
  }
  shell_energy_finalize<<<1, 256, 0, stream>>>(
      partials, nblocks, 1.0f / (float)batch, (float*)d_out);
}